// GeneralConvMinAttention_58385785422517
// MI455X (gfx1250) — compile-verified
//
#include <hip/hip_runtime.h>
#include <stdint.h>

// ---------------------------------------------------------------------------
// GeneralConv (aggr='min', additive attention) for MI455X / gfx1250, wave32.
//   y_b   = x @ W_msg + (b_msg + b_edge)                [N,D]  (fp32 WMMA)
//   t[i]  = dot(y_b[i], att_msg)                        [N]
//   c1    = dot(W_edge, att_msg)                        scalar
//   logit = leakyrelu(t[src] + c1*edge_attr, 0.2)       [E]
//   softmax by dst (flipped-uint atomicMax + f32 atomicAdd)
//   msg   = (y_b[src] + edge_attr*W_edge) * alpha
//   agg   = segment_min(msg, dst)  via flipped-uint atomicMin; inf/empty -> 0
//   out   = agg + x
// ---------------------------------------------------------------------------

#define DIMS 128

typedef __attribute__((ext_vector_type(2))) float v2f;
typedef __attribute__((ext_vector_type(8))) float v8f;

// Order-preserving float -> uint map (min/max of floats == min/max of uints)
__device__ __forceinline__ unsigned flipf(float f) {
  unsigned u = __float_as_uint(f);
  return (u & 0x80000000u) ? ~u : (u | 0x80000000u);
}
__device__ __forceinline__ unsigned unflipu(unsigned v) {
  return (v & 0x80000000u) ? (v ^ 0x80000000u) : ~v;
}

// ---------------- init sentinels ----------------
__global__ void init_ws_kernel(unsigned* __restrict__ agg_u,
                               unsigned* __restrict__ gmax_u,
                               float* __restrict__ denom,
                               long total, int nNodes) {
  long i = (long)blockIdx.x * blockDim.x + threadIdx.x;
  if (i < total) agg_u[i] = 0xFFFFFFFFu;          // flipped "+inf and beyond"
  if (i < nNodes) { gmax_u[i] = 0u; denom[i] = 0.0f; }
}

// ---------------- c1 = dot(W_edge, att) ----------------
__global__ void compute_c1_kernel(const float* __restrict__ Wedge,
                                  const float* __restrict__ att,
                                  float* __restrict__ c1) {
  __shared__ float red[DIMS];
  int tid = threadIdx.x;
  red[tid] = Wedge[tid] * att[tid];
  __syncthreads();
  for (int s = DIMS / 2; s > 0; s >>= 1) {
    if (tid < s) red[tid] += red[tid + s];
    __syncthreads();
  }
  if (tid == 0) *c1 = red[0];
}

// ---------------- y_b = x @ W_msg + bias  (fp32 WMMA 16x16x4) ----------------
// One block per 16-row stripe; 8 waves each own one 16-col tile.
// W_msg (64 KB) staged in LDS (320 KB/WGP available on CDNA5).
__global__ __launch_bounds__(256) void gemm_wmma_kernel(
    const float* __restrict__ x, const float* __restrict__ Wmsg,
    const float* __restrict__ bmsg, const float* __restrict__ bedge,
    float* __restrict__ yb) {
  __shared__ float Wl[DIMS * DIMS];
  {
    const float4* Wv = (const float4*)Wmsg;
    float4* Wlv = (float4*)Wl;
    for (int i = threadIdx.x; i < (DIMS * DIMS) / 4; i += 256) Wlv[i] = Wv[i];
  }
  __syncthreads();

  const int wave  = threadIdx.x >> 5;   // 0..7 -> column tile
  const int lane  = threadIdx.x & 31;
  const int lhalf = lane >> 4;          // 0: K+0/1 rows, 1: K+2/3 rows
  const int lmod  = lane & 15;
  const int rowTile = blockIdx.x;       // N % 16 == 0 (50000/16 = 3125)
  const int row = rowTile * 16 + lmod;  // A: M = lane%16
  const int col = wave * 16 + lmod;     // B/C/D: N = lane%16

  const float2* xr = (const float2*)(x + (size_t)row * DIMS);

  v8f acc = {0.f, 0.f, 0.f, 0.f, 0.f, 0.f, 0.f, 0.f};
#pragma unroll 4
  for (int k = 0; k < DIMS; k += 4) {
    const int ka = k + lhalf * 2;       // this half-wave's K pair
    float2 a2 = xr[ka >> 1];            // A: x[row][ka], x[row][ka+1]
    v2f av; av.x = a2.x; av.y = a2.y;
    v2f bv;                              // B: W[ka][col], W[ka+1][col]
    bv.x = Wl[ka * DIMS + col];
    bv.y = Wl[(ka + 1) * DIMS + col];
    acc = __builtin_amdgcn_wmma_f32_16x16x4_f32(
        /*neg_a=*/false, av, /*neg_b=*/false, bv,
        /*c_mod=*/(short)0, acc, /*reuse_a=*/false, /*reuse_b=*/false);
  }

  const float bias = bmsg[col] + bedge[col];
#pragma unroll
  for (int v = 0; v < 8; ++v) {         // C/D: VGPR v -> M = v + 8*lhalf
    const int m = rowTile * 16 + v + lhalf * 8;
    yb[(size_t)m * DIMS + col] = acc[v] + bias;
  }
}

// ---------------- t[i] = dot(y_b[i], att)  (one wave per node) ----------------
__global__ __launch_bounds__(256) void node_logit_kernel(
    const float* __restrict__ yb, const float* __restrict__ att,
    float* __restrict__ t, int nNodes) {
  const int wave = threadIdx.x >> 5;
  const int lane = threadIdx.x & 31;
  const int node = blockIdx.x * 8 + wave;
  if (node >= nNodes) return;
  const float* r = yb + (size_t)node * DIMS;
  float s = 0.f;
#pragma unroll
  for (int j = 0; j < 4; ++j) s += r[lane + j * 32] * att[lane + j * 32];
#pragma unroll
  for (int off = 16; off > 0; off >>= 1) s += __shfl_xor(s, off, 32);
  if (lane == 0) t[node] = s;
}

// ---------------- edge pass 1: logits + segment_max ----------------
__global__ void edge_pass1_kernel(const int* __restrict__ ei,
                                  const float* __restrict__ eattr,
                                  const float* __restrict__ t,
                                  const float* __restrict__ c1p,
                                  float* __restrict__ logits,
                                  unsigned* __restrict__ gmax_u, int E) {
  int e = blockIdx.x * blockDim.x + threadIdx.x;
  if (e >= E) return;
  int s = ei[e], d = ei[E + e];
  float v = t[s] + (*c1p) * eattr[e];
  v = v > 0.f ? v : 0.2f * v;                       // leaky_relu(0.2)
  logits[e] = v;
  atomicMax(&gmax_u[d], flipf(v));
}

// ---------------- edge pass 2: exp + segment_sum ----------------
__global__ void edge_pass2_kernel(const int* __restrict__ ei,
                                  const float* __restrict__ logits,
                                  const unsigned* __restrict__ gmax_u,
                                  float* __restrict__ eabuf,
                                  float* __restrict__ denom, int E) {
  int e = blockIdx.x * blockDim.x + threadIdx.x;
  if (e >= E) return;
  int d = ei[E + e];
  float g = __uint_as_float(unflipu(gmax_u[d]));    // finite for any dst w/ edges
  float ea = __expf(logits[e] - g);
  eabuf[e] = ea;
  atomicAdd(&denom[d], ea);
}

// ---------------- edge pass 3: msg + segment_min (one wave per edge) ----------
__global__ __launch_bounds__(256) void edge_pass3_kernel(
    const int* __restrict__ ei, const float* __restrict__ eattr,
    const float* __restrict__ yb, const float* __restrict__ Wedge,
    const float* __restrict__ eabuf, const float* __restrict__ denom,
    unsigned* __restrict__ agg_u, int E) {
  long g = (long)blockIdx.x * blockDim.x + threadIdx.x;
  int e = (int)(g >> 5);
  if (e >= E) return;
  const int lane = (int)(g & 31);
  const int s = ei[e], d = ei[E + e];
  const float alpha = eabuf[e] / (denom[d] + 1e-16f);
  const float w = eattr[e];
  const int base = lane * 4;
  const float4 y4 = *(const float4*)(yb + (size_t)s * DIMS + base);
  const float4 we = *(const float4*)(Wedge + base);
  unsigned* dsta = agg_u + (size_t)d * DIMS + base;
  atomicMin(dsta + 0, flipf((y4.x + w * we.x) * alpha));
  atomicMin(dsta + 1, flipf((y4.y + w * we.y) * alpha));
  atomicMin(dsta + 2, flipf((y4.z + w * we.z) * alpha));
  atomicMin(dsta + 3, flipf((y4.w + w * we.w) * alpha));
}

// ---------------- finalize: guard inf/empty, add residual ----------------
__global__ void finalize_kernel(const unsigned* __restrict__ agg_u,
                                const float* __restrict__ x,
                                float* __restrict__ out, long total) {
  long i = (long)blockIdx.x * blockDim.x + threadIdx.x;
  if (i >= total) return;
  unsigned uf = unflipu(agg_u[i]);
  bool finite = (uf & 0x7F800000u) != 0x7F800000u;  // empty sentinel -> NaN bits
  out[i] = (finite ? __uint_as_float(uf) : 0.0f) + x[i];
}

// ---------------------------------------------------------------------------
extern "C" void kernel_launch(void* const* d_in, const int* in_sizes, int n_in,
                              void* d_out, int out_size, void* d_ws, size_t ws_size,
                              hipStream_t stream) {
  const float* x     = (const float*)d_in[0];   // [N,128]
  const int*   ei    = (const int*)  d_in[1];   // [2,E]
  const float* eattr = (const float*)d_in[2];   // [E]
  const float* Wmsg  = (const float*)d_in[3];   // [128,128]
  const float* bmsg  = (const float*)d_in[4];   // [128]
  const float* Wedge = (const float*)d_in[5];   // [1,128]
  const float* bedge = (const float*)d_in[6];   // [128]
  const float* att   = (const float*)d_in[7];   // [128]
  float* out = (float*)d_out;

  const int nNodes = in_sizes[0] / DIMS;        // 50000 (multiple of 16)
  const int E      = in_sizes[2];               // 800000
  const size_t ND  = (size_t)nNodes * DIMS;

  // workspace layout (fp32 slots, 16B-friendly since N,D multiples of 4)
  float*    yb     = (float*)d_ws;              // [N*D]
  unsigned* agg_u  = (unsigned*)(yb + ND);      // [N*D]
  float*    t      = (float*)(agg_u + ND);      // [N]
  unsigned* gmax_u = (unsigned*)(t + nNodes);   // [N]
  float*    denom  = (float*)(gmax_u + nNodes); // [N]
  float*    logits = denom + nNodes;            // [E]
  float*    eabuf  = logits + E;                // [E]
  float*    c1     = eabuf + E;                 // [1]

  init_ws_kernel<<<(int)((ND + 255) / 256), 256, 0, stream>>>(
      agg_u, gmax_u, denom, (long)ND, nNodes);
  compute_c1_kernel<<<1, DIMS, 0, stream>>>(Wedge, att, c1);
  gemm_wmma_kernel<<<nNodes / 16, 256, 0, stream>>>(x, Wmsg, bmsg, bedge, yb);
  node_logit_kernel<<<(nNodes + 7) / 8, 256, 0, stream>>>(yb, att, t, nNodes);
  edge_pass1_kernel<<<(E + 255) / 256, 256, 0, stream>>>(
      ei, eattr, t, c1, logits, gmax_u, E);
  edge_pass2_kernel<<<(E + 255) / 256, 256, 0, stream>>>(
      ei, logits, gmax_u, eabuf, denom, E);
  edge_pass3_kernel<<<(int)(((long)E * 32 + 255) / 256), 256, 0, stream>>>(
      ei, eattr, yb, Wedge, eabuf, denom, agg_u, E);
  finalize_kernel<<<(int)((ND + 255) / 256), 256, 0, stream>>>(
      agg_u, x, out, (long)ND);
}